// Attention2D_1915555414261
// MI455X (gfx1250) — compile-verified
//
#include <hip/hip_runtime.h>

// ---------------- problem constants ----------------
#define BATCH   64
#define CIN     512
#define HID     512
#define HH      28
#define WW      28
#define HWP     784        // 28*28
#define KTAPS   9
#define M_TILE  64         // positions per workgroup
#define NTILE_M 13         // ceil(784/64)
#define CB      8          // input channels staged per K-block
#define CPAD    9          // padded channel stride in LDS (bank-conflict avoidance)
#define SLAB    (6 * 30 * CPAD)   // one staged slab: 6 rows x 30 cols x CPAD ch

typedef float v2f __attribute__((ext_vector_type(2)));
typedef float v8f __attribute__((ext_vector_type(8)));

// ---- CDNA5 async global->LDS copy (ASYNCcnt-tracked, no VGPR round trip) ----
__device__ __forceinline__ void async_copy_b32(const float* gsrc, unsigned lds_byte_off) {
  // VDST = VGPR holding LDS byte offset, VADDR = 64-bit global address, SADDR = off (GV mode)
  asm volatile("global_load_async_to_lds_b32 %0, %1, off"
               :
               : "v"(lds_byte_off), "v"((unsigned long long)(uintptr_t)gsrc)
               : "memory");
}

__device__ __forceinline__ void wait_async0() {
#if __has_builtin(__builtin_amdgcn_s_wait_asynccnt)
  __builtin_amdgcn_s_wait_asynccnt(0);
#else
  asm volatile("s_wait_asynccnt 0x0" ::: "memory");
#endif
}

// ---------------------------------------------------------------------------
// Kernel A: g_em[b][h] = img_fvec[b] . W1[h] + b1[h]      (tiny: 33 MFLOP)
// ---------------------------------------------------------------------------
__global__ void fc1_kernel(const float* __restrict__ img,
                           const float* __restrict__ W1,
                           const float* __restrict__ b1,
                           float* __restrict__ gem) {
  __shared__ float sx[CIN];
  const int b = blockIdx.x, tid = threadIdx.x;
  for (int i = tid; i < CIN; i += 256) sx[i] = img[b * CIN + i];
  __syncthreads();
  for (int h = tid; h < HID; h += 256) {
    const float4* wr = (const float4*)(W1 + (size_t)h * CIN);
    const float4* xr = (const float4*)sx;
    float acc = 0.f;
#pragma unroll 4
    for (int k = 0; k < CIN / 4; ++k) {
      float4 w = wr[k], x = xr[k];
      acc += w.x * x.x + w.y * x.y + w.z * x.z + w.w * x.w;
    }
    gem[b * HID + h] = acc + b1[h];
  }
}

// ---------------------------------------------------------------------------
// Kernel B: fused 3x3 conv (9 shifted GEMMs, fp32 WMMA 16x16x4)
//           + conv_b + g_em broadcast + tanh + dot(W2)  ->  logits[b][p]
// Double-buffered LDS input staging via global_load_async_to_lds_b32.
// Workgroup = 8 wave32; tile = 64 positions x 512 channels.
// ---------------------------------------------------------------------------
__launch_bounds__(256)
__global__ void conv_attn_kernel(const float* __restrict__ pf,     // [B,CIN,28,28]
                                 const float* __restrict__ cw,     // [HID,CIN,3,3]
                                 const float* __restrict__ cbias,  // [HID]
                                 const float* __restrict__ gem,    // [B,HID]
                                 const float* __restrict__ W2,     // [HID]
                                 float* __restrict__ logits) {     // [B,784]
  __shared__ float sIn[2][SLAB];     // double-buffered staged input slabs
  __shared__ float sLogit[M_TILE];

  const int tile = blockIdx.x;
  const int b    = blockIdx.y;
  const int p0   = tile * M_TILE;
  const int y0   = p0 / WW;
  const int tid  = threadIdx.x;
  const int lane = tid & 31;
  const int wv   = tid >> 5;         // wave id 0..7
  const int nlow = lane & 15;        // N (or A-row) index within fragment
  const int kh   = lane >> 4;        // selects K pair {0,1} vs {2,3}
  const int n_base = wv * 64;

  // Zero both buffers once: async copies below only overwrite in-bounds
  // elements, whose pattern is identical for every channel block, so the
  // conv halo / out-of-image entries stay zero for the whole kernel.
  for (int i = tid; i < 2 * SLAB; i += 256) ((float*)sIn)[i] = 0.f;
  if (tid < M_TILE) sLogit[tid] = 0.f;
  __syncthreads();

  // Per-lane A-row geometry for the 4 M sub-tiles (A frag rows = lane&15).
  int mrow[4], mcol[4];
#pragma unroll
  for (int mi = 0; mi < 4; ++mi) {
    int p = p0 + mi * 16 + nlow;
    mrow[mi] = p / WW - y0;          // row base within staged 6-row slab
    mcol[mi] = p % WW;               // x coordinate
  }

  v8f acc[4][4];
#pragma unroll
  for (int mi = 0; mi < 4; ++mi)
#pragma unroll
    for (int ni = 0; ni < 4; ++ni) acc[mi][ni] = (v8f)0.f;

  const int c0l = 2 * kh;            // this lane's channel-pair base in a K step

  // ---- async-stage channel block 0 into buffer 0 ----
  {
    const int cblk = 0;
    for (int i = tid; i < 6 * 30 * CB; i += 256) {
      int c   = i / 180;
      int rem = i - c * 180;
      int row = rem / 30;
      int col = rem - row * 30;
      int gy = y0 - 1 + row;
      int gx = col - 1;
      if ((unsigned)gy < HH && (unsigned)gx < WW)
        async_copy_b32(&pf[(((size_t)b * CIN + (cblk + c)) * HH + gy) * WW + gx],
                       (unsigned)(uintptr_t)&sIn[0][rem * CPAD + c]);
    }
  }
  wait_async0();
  __syncthreads();

  int cur = 0;
  for (int cblk = 0; cblk < CIN; cblk += CB) {
    // ---- issue async staging of the NEXT block into the other buffer ----
    if (cblk + CB < CIN) {
      float* dst = sIn[cur ^ 1];
      for (int i = tid; i < 6 * 30 * CB; i += 256) {
        int c   = i / 180;
        int rem = i - c * 180;
        int row = rem / 30;
        int col = rem - row * 30;
        int gy = y0 - 1 + row;
        int gx = col - 1;
        if ((unsigned)gy < HH && (unsigned)gx < WW)
          async_copy_b32(&pf[(((size_t)b * CIN + (cblk + CB + c)) * HH + gy) * WW + gx],
                         (unsigned)(uintptr_t)&dst[rem * CPAD + c]);
      }
      // prefetch next weight slice into L2 (global_prefetch_b8)
      int n = (tid * 2) & (HID - 1);
      __builtin_prefetch(&cw[(size_t)n * (CIN * KTAPS) + (size_t)(cblk + CB) * KTAPS], 0, 1);
    }

    // ---- 9 taps x 2 channel quads = 18 K-steps of the fp32 WMMA ----
    const float* sbuf = sIn[cur];
    for (int t = 0; t < KTAPS; ++t) {
      int r = t / 3, s = t - 3 * r;
#pragma unroll
      for (int cs = 0; cs < 2; ++cs) {
        int cl = cs * 4 + c0l;       // channel offset in block for this lane's pair
        // A fragments (LDS): A[m][k] = input[c][y+r-1][x+s-1]
        v2f a[4];
#pragma unroll
        for (int mi = 0; mi < 4; ++mi) {
          int idx = ((mrow[mi] + r) * 30 + (mcol[mi] + s)) * CPAD + cl;
          a[mi].x = sbuf[idx];
          a[mi].y = sbuf[idx + 1];
        }
        // B fragments (global, L2-resident): B[k][n] = conv_w[n][c][r][s]
        v2f bf[4];
#pragma unroll
        for (int ni = 0; ni < 4; ++ni) {
          int n = n_base + ni * 16 + nlow;
          const float* wp = cw + (size_t)n * (CIN * KTAPS)
                               + (size_t)(cblk + cl) * KTAPS + t;
          bf[ni].x = wp[0];
          bf[ni].y = wp[KTAPS];      // next channel, same tap
        }
#pragma unroll
        for (int mi = 0; mi < 4; ++mi)
#pragma unroll
          for (int ni = 0; ni < 4; ++ni)
            acc[mi][ni] = __builtin_amdgcn_wmma_f32_16x16x4_f32(
                false, a[mi], false, bf[ni], (short)0, acc[mi][ni], false, false);
      }
    }

    // next block's async copies have been overlapping the WMMAs above
    wait_async0();
    __syncthreads();
    cur ^= 1;
  }

  // ---- epilogue: + conv_b + g_em, tanh, * W2, reduce over HID ----
  float cbv[4], gev[4], w2v[4];
#pragma unroll
  for (int ni = 0; ni < 4; ++ni) {
    int n = n_base + ni * 16 + nlow;
    cbv[ni] = cbias[n];
    gev[ni] = gem[b * HID + n];
    w2v[ni] = W2[n];
  }
#pragma unroll
  for (int mi = 0; mi < 4; ++mi) {
#pragma unroll
    for (int j = 0; j < 8; ++j) {    // C layout: VGPR j -> M = j (lanes<16) / j+8
      float s = 0.f;
#pragma unroll
      for (int ni = 0; ni < 4; ++ni) {
        float v = acc[mi][ni][j] + cbv[ni] + gev[ni];
        s += tanhf(v) * w2v[ni];
      }
      // reduce over the 16 N-lanes of each half-wave
      s += __shfl_xor(s, 1, 32);
      s += __shfl_xor(s, 2, 32);
      s += __shfl_xor(s, 4, 32);
      s += __shfl_xor(s, 8, 32);
      if (nlow == 0) {
        int m = mi * 16 + j + 8 * kh;
        atomicAdd(&sLogit[m], s);    // ds_add_f32 across the 8 waves
      }
    }
  }
  __syncthreads();
  if (tid < M_TILE && (p0 + tid) < HWP)
    logits[b * HWP + p0 + tid] = sLogit[tid];
}

// ---------------------------------------------------------------------------
// Kernel C: softmax over 784 positions + attention-weighted channel sum
// ---------------------------------------------------------------------------
__global__ void softmax_attn_kernel(const float* __restrict__ pf,
                                    const float* __restrict__ logits,
                                    const float* __restrict__ b2,
                                    float* __restrict__ out) {
  __shared__ float w[HWP];
  __shared__ float red[256];
  const int b = blockIdx.x, tid = threadIdx.x;

  float lmax = -1e30f;
  for (int p = tid; p < HWP; p += 256) {
    float v = logits[b * HWP + p] + b2[0];
    w[p] = v;
    lmax = fmaxf(lmax, v);
  }
  red[tid] = lmax;
  __syncthreads();
  for (int off = 128; off > 0; off >>= 1) {
    if (tid < off) red[tid] = fmaxf(red[tid], red[tid + off]);
    __syncthreads();
  }
  float mx = red[0];
  __syncthreads();

  float lsum = 0.f;
  for (int p = tid; p < HWP; p += 256) {
    float e = expf(w[p] - mx);
    w[p] = e;
    lsum += e;
  }
  red[tid] = lsum;
  __syncthreads();
  for (int off = 128; off > 0; off >>= 1) {
    if (tid < off) red[tid] += red[tid + off];
    __syncthreads();
  }
  float inv = 1.f / red[0];
  __syncthreads();

  for (int c = tid; c < CIN; c += 256) {
    const float4* row = (const float4*)(pf + ((size_t)b * CIN + c) * HWP);
    const float4* wv4 = (const float4*)w;
    float acc = 0.f;
#pragma unroll 4
    for (int q = 0; q < HWP / 4; ++q) {
      float4 rv = row[q], wv = wv4[q];
      acc += rv.x * wv.x + rv.y * wv.y + rv.z * wv.z + rv.w * wv.w;
    }
    out[b * CIN + c] = acc * inv;
  }
}

// ---------------------------------------------------------------------------
extern "C" void kernel_launch(void* const* d_in, const int* in_sizes, int n_in,
                              void* d_out, int out_size, void* d_ws, size_t ws_size,
                              hipStream_t stream) {
  const float* img   = (const float*)d_in[0];  // [64,512]
  const float* pf    = (const float*)d_in[1];  // [64,512,28,28]
  const float* W1    = (const float*)d_in[2];  // [512,512]
  const float* b1    = (const float*)d_in[3];  // [512]
  const float* cw    = (const float*)d_in[4];  // [512,512,3,3]
  const float* cbias = (const float*)d_in[5];  // [512]
  const float* W2    = (const float*)d_in[6];  // [1,512]
  const float* b2    = (const float*)d_in[7];  // [1]
  float* out = (float*)d_out;                  // [64,512]

  float* gem    = (float*)d_ws;                // 64*512 floats
  float* logits = gem + BATCH * HID;           // 64*784 floats

  fc1_kernel<<<BATCH, 256, 0, stream>>>(img, W1, b1, gem);
  dim3 gB(NTILE_M, BATCH);
  conv_attn_kernel<<<gB, 256, 0, stream>>>(pf, cw, cbias, gem, W2, logits);
  softmax_attn_kernel<<<BATCH, 256, 0, stream>>>(pf, logits, b2, out);
}